// FeatureNestedMatryoshkaTXCDR_77266461655439
// MI455X (gfx1250) — compile-verified
//
#include <hip/hip_runtime.h>
#include <hip/hip_bf16.h>

// Problem constants (match reference)
#define BATCH 4096
#define TT 8
#define DIN 768
#define KDIM (TT * DIN)      // 6144 : flattened (t,d) contraction dim
#define DSAE 8192
#define TOPK 64

#define KBLK 16              // K-chunk staged per LDS buffer
#define NPANEL 256           // N columns per block (8 waves x 2 tiles x 16)

typedef __attribute__((ext_vector_type(2))) float v2f;
typedef __attribute__((ext_vector_type(8))) float v8f;

// ---- CDNA5 async global->LDS copy (ASYNCcnt-tracked), per ISA 15.18.3 ------
__device__ __forceinline__ void async_g2l_b128(unsigned lds_byte_off,
                                               const float* gsrc) {
    // VDST = LDS byte address, VADDR = 64-bit global address
    asm volatile("global_load_async_to_lds_b128 %0, %1, off"
                 :: "v"(lds_byte_off), "v"(gsrc)
                 : "memory");
}
__device__ __forceinline__ void wait_asynccnt0() {
    asm volatile("s_wait_asynccnt 0x0" ::: "memory");
}
__device__ __forceinline__ unsigned lds_off(const void* p) {
    // LDS aperture: low 32 bits of a generic LDS pointer are the wave-relative
    // LDS byte offset (ISA 10.2 aperture mapping).
    return (unsigned)(unsigned long long)p;
}

// ---------------------------------------------------------------------------
// Kernel 1: encoder GEMM  pre[b][s] = sum_k x[b][k] * W[k][s] + b_enc[s]
// fp32 WMMA 16x16x4. Block = 256 threads = 8 waves. Block tile = 16(M) x
// 256(N); each wave owns two 16x16 accumulators (independent WMMA chains).
// A (16xKBLK) and B (KBLKx256) panels are staged into double-buffered LDS
// with GLOBAL_LOAD_ASYNC_TO_LDS_B128 so the copy of chunk i+1 overlaps the
// WMMA burst on chunk i. Grid = (8192/256, 4096/16).
// ---------------------------------------------------------------------------
__global__ __launch_bounds__(256)
void enc_gemm_wmma(const float* __restrict__ x, const float* __restrict__ W,
                   const float* __restrict__ be, float* __restrict__ pre) {
    __shared__ __align__(16) float lds_a[2][16 * KBLK];      //  2 KB
    __shared__ __align__(16) float lds_b[2][KBLK * NPANEL];  // 32 KB

    const int tid  = threadIdx.x;
    const int lane = tid & 31;
    const int wave = tid >> 5;
    const int m0   = blockIdx.y * 16;
    const int nblk = blockIdx.x * NPANEL;
    const int half = lane >> 4;        // 0: K pair {0,1}, 1: K pair {2,3}
    const int mrow = lane & 15;        // M for A, N for B/C/D
    const int nw   = wave * 32;        // this wave's two tiles: nw, nw+16

    // --- async stage of one K-chunk into buffer `buf` -----------------------
    auto stage = [&](int buf, int k0) {
        // A: 16 rows x KBLK floats = 64 b128 chunks -> threads 0..63
        if (tid < 64) {
            const int r = tid >> 2;            // 0..15
            const int q = (tid & 3) << 2;      // 0,4,8,12
            async_g2l_b128(lds_off(&lds_a[buf][r * KBLK + q]),
                           x + (size_t)(m0 + r) * KDIM + k0 + q);
        }
        // B: KBLK rows x 256 floats = 1024 b128 chunks -> 4 per thread
#pragma unroll
        for (int j = 0; j < 4; ++j) {
            const int c    = tid + j * 256;    // 0..1023
            const int kr   = c >> 6;           // 0..15
            const int colq = (c & 63) << 2;    // 0..252
            async_g2l_b128(lds_off(&lds_b[buf][kr * NPANEL + colq]),
                           W + (size_t)(k0 + kr) * DSAE + nblk + colq);
        }
    };

    v8f c0 = {}, c1 = {};

    stage(0, 0);
    for (int it = 0; it < KDIM / KBLK; ++it) {
        const int buf = it & 1;
        wait_asynccnt0();
        __syncthreads();                       // buffer `buf` ready
        if (it + 1 < KDIM / KBLK) stage(buf ^ 1, (it + 1) * KBLK);

#pragma unroll
        for (int kk = 0; kk < KBLK; kk += 4) {
            // A float2 (ds_load_b64): row mrow, K = kk + 2*half, +1
            v2f a = *(const v2f*)&lds_a[buf][mrow * KBLK + kk + 2 * half];
            // B: VGPR0 -> K = kk+2*half, VGPR1 -> K = kk+2*half+1
            const int kb = (kk + 2 * half) * NPANEL;
            v2f b0, b1;
            b0.x = lds_b[buf][kb + nw + mrow];
            b0.y = lds_b[buf][kb + NPANEL + nw + mrow];
            b1.x = lds_b[buf][kb + nw + 16 + mrow];
            b1.y = lds_b[buf][kb + NPANEL + nw + 16 + mrow];
            c0 = __builtin_amdgcn_wmma_f32_16x16x4_f32(
                false, a, false, b0, (short)0, c0, false, false);
            c1 = __builtin_amdgcn_wmma_f32_16x16x4_f32(
                false, a, false, b1, (short)0, c1, false, false);
        }
        __syncthreads();                       // all reads of `buf` done
    }

    const float bias0 = be[nblk + nw + mrow];
    const float bias1 = be[nblk + nw + 16 + mrow];
    // C/D layout: VGPR r -> M = r + 8*half, N = mrow
    float* out0 = pre + (size_t)(m0 + 8 * half) * DSAE + (nblk + nw + mrow);
#pragma unroll
    for (int r = 0; r < 8; ++r) {
        out0[(size_t)r * DSAE]      = c0[r] + bias0;
        out0[(size_t)r * DSAE + 16] = c1[r] + bias1;
    }
}

// ---------------------------------------------------------------------------
// Kernel 2: per-row top-64 (stable: ties -> smallest index, like lax.top_k),
// then rewrite the row as sparse relu(z). One block per row, row cached in
// registers (32 f32 / thread), 64 x (local scan + LDS tree argmax).
// ---------------------------------------------------------------------------
__global__ __launch_bounds__(256)
void topk_sparsify(float* __restrict__ z, int* __restrict__ oidx,
                   float* __restrict__ ovals) {
    __shared__ float sval[256];
    __shared__ int   sidx[256];
    __shared__ float wval[TOPK];
    __shared__ int   widx[TOPK];

    const int tid = threadIdx.x;
    const int b   = blockIdx.x;
    float* row = z + (size_t)b * DSAE;

    float v[32];
#pragma unroll
    for (int j = 0; j < 32; ++j) v[j] = row[tid + j * 256];

    for (int it = 0; it < TOPK; ++it) {
        // local argmax; strict '>' keeps smallest j (== smallest column)
        float bv = v[0]; int bj = 0;
#pragma unroll
        for (int j = 1; j < 32; ++j)
            if (v[j] > bv) { bv = v[j]; bj = j; }
        sval[tid] = bv;
        sidx[tid] = tid + bj * 256;
        __syncthreads();
        for (int s = 128; s > 0; s >>= 1) {
            if (tid < s) {
                const float ov = sval[tid + s];
                const int   oi = sidx[tid + s];
                if (ov > sval[tid] || (ov == sval[tid] && oi < sidx[tid])) {
                    sval[tid] = ov; sidx[tid] = oi;
                }
            }
            __syncthreads();
        }
        const float win_v = sval[0];
        const int   win_i = sidx[0];
        if (tid == 0) { wval[it] = win_v; widx[it] = win_i; }
        if ((win_i & 255) == tid) v[win_i >> 8] = -__builtin_inff();
        __syncthreads();
    }

    // dense zero, then scatter relu'd winners
#pragma unroll
    for (int j = 0; j < 32; ++j) row[tid + j * 256] = 0.0f;
    __syncthreads();
    if (tid < TOPK) {
        const float r = wval[tid] > 0.0f ? wval[tid] : 0.0f;
        row[widx[tid]] = r;
        oidx [b * TOPK + tid] = widx[tid];
        ovals[b * TOPK + tid] = r;
    }
}

// ---------------------------------------------------------------------------
// Kernel 3: matryoshka decode + squared-error partials.
// Block = (scale, batch-row); sparse gather-axpy over <=64 W_dec rows
// (coalesced 6144-wide rows), uniform prefix branch (index lives in LDS ->
// scalar condition, no divergence). Scale 3 also emits x_hat.
// ---------------------------------------------------------------------------
__global__ __launch_bounds__(256)
void decode_loss(const float* __restrict__ x,
                 const int* __restrict__ oidx, const float* __restrict__ ovals,
                 const float* __restrict__ Wd0, const float* __restrict__ Wd1,
                 const float* __restrict__ Wd2, const float* __restrict__ Wd3,
                 const float* __restrict__ bd0, const float* __restrict__ bd1,
                 const float* __restrict__ bd2, const float* __restrict__ bd3,
                 float* __restrict__ xhat, float* __restrict__ partials) {
    const int scale = blockIdx.x;
    const int b     = blockIdx.y;
    const int tid   = threadIdx.x;

    const float* Wd = (scale == 0) ? Wd0 : (scale == 1) ? Wd1
                    : (scale == 2) ? Wd2 : Wd3;
    const float* bd = (scale == 0) ? bd0 : (scale == 1) ? bd1
                    : (scale == 2) ? bd2 : bd3;
    const int p = (scale + 1) * 2048;

    __shared__ int   fidx[TOPK];
    __shared__ float fval[TOPK];
    if (tid < TOPK) {
        fidx[tid] = oidx [b * TOPK + tid];
        fval[tid] = ovals[b * TOPK + tid];
    }
    __syncthreads();

    float acc[24];
#pragma unroll
    for (int j = 0; j < 24; ++j) acc[j] = bd[tid + j * 256];

    for (int k = 0; k < TOPK; ++k) {
        const int fi = fidx[k];
        if (fi < p) {
            const float v  = fval[k];
            const float* wr = Wd + (size_t)fi * KDIM;
#pragma unroll
            for (int j = 0; j < 24; ++j)
                acc[j] = fmaf(v, wr[tid + j * 256], acc[j]);
        }
    }

    const float* xr = x + (size_t)b * KDIM;
    float s = 0.0f;
#pragma unroll
    for (int j = 0; j < 24; ++j) {
        const float d = acc[j] - xr[tid + j * 256];
        s += d * d;
        if (scale == 3) xhat[(size_t)b * KDIM + tid + j * 256] = acc[j];
    }

    __shared__ float red[256];
    red[tid] = s;
    __syncthreads();
    for (int st = 128; st > 0; st >>= 1) {
        if (tid < st) red[tid] += red[tid + st];
        __syncthreads();
    }
    if (tid == 0) partials[b * 4 + scale] = red[0];
}

// ---------------------------------------------------------------------------
// Kernel 4: deterministic fixed-order reduction of 16384 partials -> l_recon.
// l_recon = total_sumsq / (B * n_scales)
// ---------------------------------------------------------------------------
__global__ __launch_bounds__(256)
void loss_reduce(const float* __restrict__ partials, float* __restrict__ out) {
    __shared__ float red[256];
    const int tid = threadIdx.x;
    float s = 0.0f;
    for (int i = 0; i < 64; ++i) s += partials[tid + i * 256];
    red[tid] = s;
    __syncthreads();
    for (int st = 128; st > 0; st >>= 1) {
        if (tid < st) red[tid] += red[tid + st];
        __syncthreads();
    }
    if (tid == 0) out[0] = red[0] * (1.0f / (4096.0f * 4.0f));
}

// ---------------------------------------------------------------------------
extern "C" void kernel_launch(void* const* d_in, const int* in_sizes, int n_in,
                              void* d_out, int out_size, void* d_ws, size_t ws_size,
                              hipStream_t stream) {
    (void)in_sizes; (void)n_in; (void)out_size; (void)ws_size;

    const float* x     = (const float*)d_in[0];
    const float* W_enc = (const float*)d_in[1];
    const float* b_enc = (const float*)d_in[2];
    const float* Wd0   = (const float*)d_in[3];
    const float* bd0   = (const float*)d_in[4];
    const float* Wd1   = (const float*)d_in[5];
    const float* bd1   = (const float*)d_in[6];
    const float* Wd2   = (const float*)d_in[7];
    const float* bd2   = (const float*)d_in[8];
    const float* Wd3   = (const float*)d_in[9];
    const float* bd3   = (const float*)d_in[10];

    float* out   = (float*)d_out;
    float* l_out = out;                                  // [0]          scalar
    float* xhat  = out + 1;                              // [1 ..)       B*T*DIN
    float* z     = out + 1 + (size_t)BATCH * KDIM;       // then         B*DSAE

    int*   oidx     = (int*)d_ws;                                        // 1 MB
    float* ovals    = (float*)((char*)d_ws + (size_t)BATCH * TOPK * 4);  // 1 MB
    float* partials = (float*)((char*)d_ws + 2u * (size_t)BATCH * TOPK * 4);

    // 1) pre = x @ W_enc + b_enc  (written into z slice as scratch)
    enc_gemm_wmma<<<dim3(DSAE / NPANEL, BATCH / 16), 256, 0, stream>>>(
        x, W_enc, b_enc, z);

    // 2) top-64 per row -> sparse relu z + compacted (idx, val) lists
    topk_sparsify<<<dim3(BATCH), 256, 0, stream>>>(z, oidx, ovals);

    // 3) 4-scale sparse decode + per-(row,scale) squared-error partials
    decode_loss<<<dim3(4, BATCH), 256, 0, stream>>>(
        x, oidx, ovals, Wd0, Wd1, Wd2, Wd3, bd0, bd1, bd2, bd3,
        xhat, partials);

    // 4) deterministic scalar loss
    loss_reduce<<<dim3(1), 256, 0, stream>>>(partials, l_out);
}